// CSOCSSC_V1251_5634997092634
// MI455X (gfx1250) — compile-verified
//
#include <hip/hip_runtime.h>
#include <hip/hip_bf16.h>

#define N_RES   4096
#define N_PAIRS 500000
// contact work units: 32-row x 16-col blocks
#define UNITS_I (N_RES / 32)                 // 128
#define UNITS_J (N_RES / 16)                 // 256
#define N_UNITS (UNITS_I * UNITS_J)          // 32768

typedef __attribute__((ext_vector_type(2))) float v2f;
typedef __attribute__((ext_vector_type(8))) float v8f;

__device__ __forceinline__ double wave_reduce(double v) {
    // wave32 butterfly reduction
    #pragma unroll
    for (int off = 16; off > 0; off >>= 1)
        v += __shfl_xor(v, off, 32);
    return v;
}

// ---------------------------------------------------------------------------
// Zero the accumulators: [0]=bond [1]=clash [2]=contact [3]=hb
// ---------------------------------------------------------------------------
__global__ void bk_init(double* acc) {
    if (threadIdx.x < 8) acc[threadIdx.x] = 0.0;
}

// ---------------------------------------------------------------------------
// Per-residue: squared norm, backbone N/O atoms, bond energy partial
// ---------------------------------------------------------------------------
__global__ void bk_backbone(const float* __restrict__ ca,
                            float* __restrict__ sq,
                            float* __restrict__ ncrd,
                            float* __restrict__ ocrd,
                            double* __restrict__ acc) {
    int i = blockIdx.x * blockDim.x + threadIdx.x;
    double bond = 0.0;
    if (i < N_RES) {
        float cax = ca[3*i+0], cay = ca[3*i+1], caz = ca[3*i+2];
        sq[i] = cax*cax + cay*cay + caz*caz;

        float upx = 0.f, upy = 0.f, upz = 0.f;  // unit[i-1]
        float uix = 0.f, uiy = 0.f, uiz = 0.f;  // unit[i]
        if (i > 0) {
            float vx = cax - ca[3*(i-1)+0];
            float vy = cay - ca[3*(i-1)+1];
            float vz = caz - ca[3*(i-1)+2];
            float nrm = sqrtf(vx*vx + vy*vy + vz*vz) + 1e-8f;
            upx = vx / nrm; upy = vy / nrm; upz = vz / nrm;
        }
        if (i < N_RES - 1) {
            float vx = ca[3*(i+1)+0] - cax;
            float vy = ca[3*(i+1)+1] - cay;
            float vz = ca[3*(i+1)+2] - caz;
            float n0 = sqrtf(vx*vx + vy*vy + vz*vz);
            float nrm = n0 + 1e-8f;
            uix = vx / nrm; uiy = vy / nrm; uiz = vz / nrm;
            float db = n0 - 3.8f;
            bond = (double)(db * db);
        }
        // N atom: i==0 uses unit[0], else unit[i-1]
        float nux = (i == 0) ? uix : upx;
        float nuy = (i == 0) ? uiy : upy;
        float nuz = (i == 0) ? uiz : upz;
        float nx = cax - 0.38f * nux;
        float ny = cay - 0.38f * nuy;
        float nz = caz - 0.38f * nuz;
        // C atom: i==n-1 uses unit[n-2], else unit[i]
        float cux = (i == N_RES - 1) ? upx : uix;
        float cuy = (i == N_RES - 1) ? upy : uiy;
        float cuz = (i == N_RES - 1) ? upz : uiz;
        float cx = cax + 0.38f * cux;
        float cy = cay + 0.38f * cuy;
        float cz = caz + 0.38f * cuz;
        // O atom
        float ox, oy, oz;
        if (i < N_RES - 1) {
            float vnx = nx - cax, vny = ny - cay, vnz = nz - caz;
            float vcx = cx - cax, vcy = cy - cay, vcz = cz - caz;
            float px = vcy * vnz - vcz * vny;
            float py = vcz * vnx - vcx * vnz;
            float pz = vcx * vny - vcy * vnx;
            float pn = sqrtf(px*px + py*py + pz*pz);
            if (pn > 1e-6f) {
                float inv = 1.0f / fmaxf(pn, 1e-12f);
                px *= inv; py *= inv; pz *= inv;
            }
            ox = cx + 1.24f * px; oy = cy + 1.24f * py; oz = cz + 1.24f * pz;
        } else {
            ox = cx; oy = cy + 1.24f; oz = cz;
        }
        ncrd[3*i+0] = nx; ncrd[3*i+1] = ny; ncrd[3*i+2] = nz;
        ocrd[3*i+0] = ox; ocrd[3*i+1] = oy; ocrd[3*i+2] = oz;
    }
    bond = wave_reduce(bond);
    if ((threadIdx.x & 31) == 0) atomicAdd(&acc[0], bond);
}

// ---------------------------------------------------------------------------
// Hydrogen-bond term: offsets 2..5, gaussian well on O(i)..N(i+off) distance
// ---------------------------------------------------------------------------
__global__ void bk_hb(const float* __restrict__ ncrd,
                      const float* __restrict__ ocrd,
                      double* __restrict__ acc) {
    int i = blockIdx.x * blockDim.x + threadIdx.x;
    double s = 0.0;
    if (i < N_RES) {
        float ox = ocrd[3*i+0], oy = ocrd[3*i+1], oz = ocrd[3*i+2];
        #pragma unroll
        for (int off = 2; off <= 5; ++off) {
            int j = i + off;
            if (j < N_RES) {
                float dx = ox - ncrd[3*j+0];
                float dy = oy - ncrd[3*j+1];
                float dz = oz - ncrd[3*j+2];
                float d = sqrtf(dx*dx + dy*dy + dz*dz);
                if (d > 2.5f && d < 3.5f) {
                    float t = (d - 2.95f) / 0.3f;
                    s += (double)expf(-t * t);
                }
            }
        }
    }
    s = wave_reduce(s);
    if ((threadIdx.x & 31) == 0) atomicAdd(&acc[3], s);
}

// ---------------------------------------------------------------------------
// Clash term over 500k int64 index pairs
// ---------------------------------------------------------------------------
__global__ void bk_clash(const float* __restrict__ ca,
                         const long long* __restrict__ pairs,
                         double* __restrict__ acc) {
    double s = 0.0;
    for (int p = blockIdx.x * blockDim.x + threadIdx.x; p < N_PAIRS;
         p += gridDim.x * blockDim.x) {
        long long i0 = pairs[2*p+0];
        long long i1 = pairs[2*p+1];
        float dx = ca[3*i0+0] - ca[3*i1+0];
        float dy = ca[3*i0+1] - ca[3*i1+1];
        float dz = ca[3*i0+2] - ca[3*i1+2];
        float d2 = dx*dx + dy*dy + dz*dz;
        float d = sqrtf(fmaxf(d2, 1e-12f));
        float r = fmaxf(3.2f - d, 0.0f);
        s += (double)(r * r);
    }
    s = wave_reduce(s);
    if ((threadIdx.x & 31) == 0) atomicAdd(&acc[1], s);
}

// ---------------------------------------------------------------------------
// Contact term: Gram matrix via V_WMMA_F32_16X16X4_F32.
// Each wave handles a 32x16 block: two stacked 16x16 tiles sharing one B
// fragment -> two WMMAs per iteration, fused sqrt+diff^2 epilogue against K.
//   D2[m][n] = sq[i0+m] + sq[j0+n] - 2 * (x_i . x_j)
// A layout (16x4 f32): lanes 0-15 {K0,K1}, lanes 16-31 {K2,K3}; B mirrors.
// C/D layout: lane<16 -> M=v, N=lane; lane>=16 -> M=v+8, N=lane-16.
// All loads are unconditional (selects, not predicated loads) so control flow
// stays uniform and EXEC is all-ones at every WMMA.
// ---------------------------------------------------------------------------
__global__ void __launch_bounds__(256)
bk_contact(const float* __restrict__ ca,
           const float* __restrict__ Kmat,
           const float* __restrict__ sq,
           double* __restrict__ acc) {
    const int  lane = threadIdx.x & 31;
    const int  l    = lane & 15;
    const bool hi   = lane >= 16;
    const int  waves_total = gridDim.x * (blockDim.x >> 5);
    const int  wid = blockIdx.x * (blockDim.x >> 5) + (threadIdx.x >> 5);
    const int  mbase = hi ? 8 : 0;

    float accf = 0.0f;
    for (int t = wid; t < N_UNITS; t += waves_total) {
        const int i0 = (t >> 8) * 32;        // 32-row block
        const int j0 = (t & 255) * 16;       // 16-col block

        // Unconditional component loads, then lane-half selects (no branches)
        const float* pa0 = ca + 3 * (i0 + l);
        const float* pa1 = ca + 3 * (i0 + 16 + l);
        const float* pb  = ca + 3 * (j0 + l);
        const float a0x = pa0[0], a0y = pa0[1], a0z = pa0[2];
        const float a1x = pa1[0], a1y = pa1[1], a1z = pa1[2];
        const float bx  = pb[0],  by  = pb[1],  bz  = pb[2];

        v2f a0, a1, b;
        a0.x = hi ? a0z : a0x;  a0.y = hi ? 0.0f : a0y;
        a1.x = hi ? a1z : a1x;  a1.y = hi ? 0.0f : a1y;
        b.x  = hi ? bz  : bx;   b.y  = hi ? 0.0f : by;

        v8f c0 = {};
        v8f c1 = {};
        c0 = __builtin_amdgcn_wmma_f32_16x16x4_f32(
                false, a0, false, b, (short)0, c0, false, false);
        c1 = __builtin_amdgcn_wmma_f32_16x16x4_f32(
                false, a1, false, b, (short)0, c1, false, false);

        // prefetch next unit's K block (global_prefetch_b8)
        const int tn = t + waves_total;
        if (tn < N_UNITS) {
            const float* pK = Kmat + (size_t)((tn >> 8) * 32) * N_RES
                                   + ((tn & 255) * 16) + l;
            __builtin_prefetch(pK, 0, 0);
        }

        const float  sqj    = sq[j0 + l];
        const float* Kbase0 = Kmat + (size_t)i0 * N_RES + j0 + l;
        const float* Kbase1 = Kbase0 + (size_t)16 * N_RES;
        #pragma unroll
        for (int v = 0; v < 8; ++v) {
            const int m = mbase + v;
            {
                float d2   = sq[i0 + m] + sqj - 2.0f * c0[v];
                float d    = sqrtf(fmaxf(d2, 1e-12f));
                float diff = d - 8.0f * (1.0f - Kbase0[(size_t)m * N_RES]);
                accf += diff * diff;
            }
            {
                float d2   = sq[i0 + 16 + m] + sqj - 2.0f * c1[v];
                float d    = sqrtf(fmaxf(d2, 1e-12f));
                float diff = d - 8.0f * (1.0f - Kbase1[(size_t)m * N_RES]);
                accf += diff * diff;
            }
        }
    }
    double s = wave_reduce((double)accf);
    if (lane == 0) atomicAdd(&acc[2], s);
}

// ---------------------------------------------------------------------------
// Combine weighted means into the final scalar
// ---------------------------------------------------------------------------
__global__ void bk_combine(const double* __restrict__ acc,
                           float* __restrict__ out) {
    if (threadIdx.x == 0 && blockIdx.x == 0) {
        double e_bond    = 30.0 * acc[0] / (double)(N_RES - 1);
        double e_clash   = 50.0 * acc[1] / (double)N_PAIRS;
        double e_contact =  5.0 * acc[2] / ((double)N_RES * (double)N_RES);
        double e_hb      =  4.0 * (-0.5 * acc[3]) / ((double)N_RES * (double)N_RES);
        out[0] = (float)(e_bond + e_clash + e_contact + e_hb);
    }
}

extern "C" void kernel_launch(void* const* d_in, const int* in_sizes, int n_in,
                              void* d_out, int out_size, void* d_ws, size_t ws_size,
                              hipStream_t stream) {
    const float*     ca    = (const float*)d_in[0];     // (4096,3) f32
    const float*     Kmat  = (const float*)d_in[1];     // (4096,4096) f32
    const long long* pairs = (const long long*)d_in[2]; // (500000,2) i64
    float*           out   = (float*)d_out;             // scalar f32

    char*   ws   = (char*)d_ws;
    double* acc  = (double*)ws;                              // 8 doubles (64 B)
    float*  sq   = (float*)(ws + 64);                        // 4096 f32
    float*  ncrd = (float*)(ws + 64 + 4096*4);               // 4096*3 f32
    float*  ocrd = (float*)(ws + 64 + 4096*4 + 4096*12);     // 4096*3 f32

    bk_init    <<<1, 32, 0, stream>>>(acc);
    bk_backbone<<<(N_RES + 255) / 256, 256, 0, stream>>>(ca, sq, ncrd, ocrd, acc);
    bk_hb      <<<(N_RES + 255) / 256, 256, 0, stream>>>(ncrd, ocrd, acc);
    bk_clash   <<<512, 256, 0, stream>>>(ca, pairs, acc);
    bk_contact <<<2048, 256, 0, stream>>>(ca, Kmat, sq, acc);
    bk_combine <<<1, 32, 0, stream>>>(acc, out);
}